// LayerTS_28355374088368
// MI455X (gfx1250) — compile-verified
//
#include <hip/hip_runtime.h>

#define IN_DIM    128
#define OUT_DIM   64
#define NUM_BASIS 8
#define SEQ       512
#define BATCH     2
#define ROWS      (BATCH * SEQ)   // 1024
#define LN_EPS    1e-5f

typedef float v2f __attribute__((ext_vector_type(2)));
typedef float v8f __attribute__((ext_vector_type(8)));

// ---------------------------------------------------------------------------
// Kernel 1: fused dual GEMM (x@M^T -> layernorm -> xn,  x@Wres^T -> residual)
// using V_WMMA_F32_16X16X4_F32 (full fp32 precision).
// grid = ROWS/16 blocks, block = 128 threads (4 waves).
// Wave w computes the 16x16 output tile for columns [16w, 16w+16).
//
// f32 WMMA operand layouts (ISA 7.12.2):
//   A (16x4, MxK):  lane L -> row m = L%16, holds K = 2*(L/16)+{0,1} in v[0],v[1]
//   B (4x16, KxN):  lane L -> col n = L%16, holds K = 2*(L/16)+{0,1} in v[0],v[1]
//   C/D (16x16):    lane L -> col n = L%16, vgpr v -> row m = v + 8*(L/16)
// ---------------------------------------------------------------------------
__global__ __launch_bounds__(128)
void k1_gemm_ln(const float* __restrict__ x,     // (ROWS, 128)
                const float* __restrict__ Mw,    // (64, 128)
                const float* __restrict__ Wres,  // (64, 128)
                const float* __restrict__ gamma, // (64,)
                const float* __restrict__ beta,  // (64,)
                float* __restrict__ xn,          // (ROWS, 64) workspace
                float* __restrict__ res_out)     // (ROWS, 64) == d_out (residual)
{
    __shared__ float ly[16][68];   // padded: bank = (4m + c) % 64, conflict-free
    __shared__ float lmu[16];
    __shared__ float lrs[16];

    const int tid  = threadIdx.x;
    const int lane = tid & 31;
    const int wave = tid >> 5;

    const int m    = lane & 15;          // row-in-tile for A / col-in-tile for B,C,D
    const int khalf = (lane >> 4) << 1;  // 0 or 2: K sub-offset for A and B fragments
    const int n0   = wave << 4;          // this wave's column tile base
    const int ncol = n0 + m;             // B-fragment column (global, 0..63)
    const int row0 = blockIdx.x << 4;    // 16 rows per block

    v8f c = {0.f,0.f,0.f,0.f,0.f,0.f,0.f,0.f};
    v8f r = {0.f,0.f,0.f,0.f,0.f,0.f,0.f,0.f};

    const float* xrow = x    + (row0 + m) * IN_DIM + khalf;  // A fragment source
    const float* mrow = Mw   + ncol * IN_DIM + khalf;        // B (M^T) source
    const float* wrow = Wres + ncol * IN_DIM + khalf;        // B (Wres^T) source

    #pragma unroll
    for (int step = 0; step < IN_DIM / 4; ++step) {
        v2f a  = *(const v2f*)(xrow + step * 4);
        v2f bm = *(const v2f*)(mrow + step * 4);
        v2f bw = *(const v2f*)(wrow + step * 4);
        c = __builtin_amdgcn_wmma_f32_16x16x4_f32(false, a, false, bm, (short)0, c, false, false);
        r = __builtin_amdgcn_wmma_f32_16x16x4_f32(false, a, false, bw, (short)0, r, false, false);
    }

    // Stage Y tiles to LDS for cross-wave layernorm statistics.
    const int msub = (lane >> 4) << 3;   // 0 or 8
    #pragma unroll
    for (int v = 0; v < 8; ++v) {
        ly[v + msub][n0 + m] = c[v];
    }
    __syncthreads();

    if (tid < 16) {
        float s = 0.f;
        #pragma unroll
        for (int j = 0; j < OUT_DIM; ++j) s += ly[tid][j];
        const float mu = s * (1.0f / OUT_DIM);
        float vv = 0.f;
        #pragma unroll
        for (int j = 0; j < OUT_DIM; ++j) {
            const float d = ly[tid][j] - mu;
            vv += d * d;
        }
        lmu[tid] = mu;
        lrs[tid] = __builtin_amdgcn_rsqf(vv * (1.0f / OUT_DIM) + LN_EPS);
    }
    __syncthreads();

    #pragma unroll
    for (int v = 0; v < 8; ++v) {
        const int mm   = v + msub;
        const int col  = n0 + m;
        const int rowg = row0 + mm;
        const float yn = (c[v] - lmu[mm]) * lrs[mm] * gamma[col] + beta[col];
        xn[rowg * OUT_DIM + col]      = yn;
        res_out[rowg * OUT_DIM + col] = r[v];   // residual parked in d_out
    }
}

// ---------------------------------------------------------------------------
// Kernel 2: basis contraction.
//   Nk[b,s,i] = sum_{j,g} P[i,j,g] * xn[b,s,j] * cos(2*pi*k_s / (i*512+j*8+g+2))
// grid = SEQ blocks (one per position), block = 256 threads.
// Thread t: i = t>>2, j-range = [(t&3)*16, +16). v_cos_f32 computes cos(2*pi*x),
// so cos(2*pi*k/p) == v_cos(fract(k * rcp(p))).
// ---------------------------------------------------------------------------
__global__ __launch_bounds__(256)
void k2_basis(const float* __restrict__ P,         // (64, 64, 8)
              const int*   __restrict__ positions, // (512,)
              const float* __restrict__ xn,        // (ROWS, 64)
              float* __restrict__ out)             // (ROWS, 64), holds residual on entry
{
    __shared__ float sx0[OUT_DIM];
    __shared__ float sx1[OUT_DIM];
    __shared__ float partial[2][OUT_DIM][4];

    const int s = blockIdx.x;
    const int t = threadIdx.x;
    const float kpos = (float)positions[s];

    if (t < OUT_DIM) {
        sx0[t] = xn[s * OUT_DIM + t];
    } else if (t < 2 * OUT_DIM) {
        sx1[t - OUT_DIM] = xn[(SEQ + s) * OUT_DIM + (t - OUT_DIM)];
    }
    __syncthreads();

    const int i  = t >> 2;
    const int jq = t & 3;
    const int j0 = jq << 4;

    const float* Pi = P + (i * OUT_DIM + j0) * NUM_BASIS;
    float acc0 = 0.f, acc1 = 0.f;

    for (int jj = 0; jj < 16; ++jj) {
        const int   j    = j0 + jj;
        const float base = (float)(i * (OUT_DIM * NUM_BASIS) + j * NUM_BASIS + 2);
        float w = 0.f;
        #pragma unroll
        for (int g = 0; g < NUM_BASIS; ++g) {
            const float p  = base + (float)g;
            const float tf = __builtin_amdgcn_fractf(kpos * __builtin_amdgcn_rcpf(p));
            const float cg = __builtin_amdgcn_cosf(tf);   // cos(2*pi*k/p)
            w = fmaf(Pi[jj * NUM_BASIS + g], cg, w);
        }
        acc0 = fmaf(w, sx0[j], acc0);
        acc1 = fmaf(w, sx1[j], acc1);
    }

    partial[0][i][jq] = acc0;
    partial[1][i][jq] = acc1;
    __syncthreads();

    if (t < OUT_DIM) {
        const float n0 = partial[0][t][0] + partial[0][t][1] + partial[0][t][2] + partial[0][t][3];
        const float n1 = partial[1][t][0] + partial[1][t][1] + partial[1][t][2] + partial[1][t][3];
        out[s * OUT_DIM + t]         = n0 + out[s * OUT_DIM + t];           // + residual
        out[(SEQ + s) * OUT_DIM + t] = n1 + out[(SEQ + s) * OUT_DIM + t];   // + residual
    }
}

// ---------------------------------------------------------------------------
extern "C" void kernel_launch(void* const* d_in, const int* in_sizes, int n_in,
                              void* d_out, int out_size, void* d_ws, size_t ws_size,
                              hipStream_t stream) {
    (void)in_sizes; (void)n_in; (void)out_size; (void)ws_size;

    const float* x         = (const float*)d_in[0];
    const int*   positions = (const int*)  d_in[1];
    const float* Mw        = (const float*)d_in[2];
    const float* P         = (const float*)d_in[3];
    const float* gamma     = (const float*)d_in[4];
    const float* beta      = (const float*)d_in[5];
    const float* Wres      = (const float*)d_in[6];

    float* out = (float*)d_out;
    float* xn  = (float*)d_ws;   // ROWS*OUT_DIM floats = 256 KB

    k1_gemm_ln<<<ROWS / 16, 128, 0, stream>>>(x, Mw, Wres, gamma, beta, xn, out);
    k2_basis<<<SEQ, 256, 0, stream>>>(P, positions, xn, out);
}